// CVRPModel_34935263986371
// MI455X (gfx1250) — compile-verified
//
#include <hip/hip_runtime.h>
#include <math.h>

typedef __attribute__((ext_vector_type(2))) float v2f;
typedef __attribute__((ext_vector_type(8))) float v8f;

#define EMB   256
#define NHEAD 8
#define HD    32
#define NODE  200
#define SOL   200
#define POMO  200
#define BATCH 128
#define NTILE 13            // ceil(200/16)
#define ROWS  (BATCH * POMO)  // 25600 global rows for all projections

// ---- workspace layout (float offsets) ----
static constexpr size_t OFF_MID = 0;                       // 24 (pad 64)
static constexpr size_t OFF_WQ  = 64;                      // 257*256 = 65792
static constexpr size_t OFF_WK  = OFF_WQ  + 65792;
static constexpr size_t OFF_WV  = OFF_WK  + 65536;
static constexpr size_t OFF_WC  = OFF_WV  + 65536;
static constexpr size_t OFF_WKS = OFF_WC  + 65536;
static constexpr size_t OFF_WVS = OFF_WKS + 65536;
static constexpr size_t OFF_Q   = OFF_WVS + 65536;
static constexpr size_t OFF_K   = OFF_Q   + (size_t)ROWS * 256;
static constexpr size_t OFF_V   = OFF_K   + (size_t)ROWS * 256;
static constexpr size_t OFF_KS  = OFF_V   + (size_t)ROWS * 256;
static constexpr size_t OFF_VS  = OFF_KS  + (size_t)ROWS * 256;
static constexpr size_t OFF_AO  = OFF_VS  + (size_t)ROWS * 256;
static constexpr size_t OFF_MH  = OFF_AO  + (size_t)ROWS * 256;

// fp32 WMMA: D(16x16) = A(16x4) * B(4x16) + C   (wave32, 8-arg VOP3P form)
__device__ __forceinline__ v8f wmma4(v2f a, v2f b, v8f c) {
  return __builtin_amdgcn_wmma_f32_16x16x4_f32(
      /*neg_a=*/false, a, /*neg_b=*/false, b,
      /*c_mod=*/(short)0, c, /*reuse_a=*/false, /*reuse_b=*/false);
}

// ---------------- tiny hypernet MLP: pref(4) -> mid(24) ----------------
__global__ __launch_bounds__(256) void mlp_kernel(
    const float* __restrict__ pref,
    const float* __restrict__ fc1w, const float* __restrict__ fc1b,
    const float* __restrict__ fc2w, const float* __restrict__ fc2b,
    const float* __restrict__ fc3w, const float* __restrict__ fc3b,
    float* __restrict__ mid) {
  __shared__ float h1[256], h2[256];
  const int i = threadIdx.x;
  const float p0 = pref[0], p1 = pref[1], p2 = pref[2], p3 = pref[3];
  h1[i] = fc1w[i * 4 + 0] * p0 + fc1w[i * 4 + 1] * p1 +
          fc1w[i * 4 + 2] * p2 + fc1w[i * 4 + 3] * p3 + fc1b[i];
  __syncthreads();
  float acc = fc2b[i];
  for (int j = 0; j < 256; ++j) acc += fc2w[i * 256 + j] * h1[j];
  h2[i] = acc;
  __syncthreads();
  if (i < 24) {
    float a = fc3b[i];
    for (int j = 0; j < 256; ++j) a += fc3w[i * 256 + j] * h2[j];
    mid[i] = a;
  }
}

// ---------------- generated weights: dst[t] = hyper[t,0:4] . mid[seg*4:..] ----
__global__ __launch_bounds__(256) void genw_kernel(
    const float* __restrict__ hyper, const float* __restrict__ mid, int seg,
    float* __restrict__ dst, int n) {
  const int t = blockIdx.x * blockDim.x + threadIdx.x;
  if (t >= n) return;
  const float m0 = mid[seg * 4 + 0], m1 = mid[seg * 4 + 1];
  const float m2 = mid[seg * 4 + 2], m3 = mid[seg * 4 + 3];
  const float4 h = ((const float4*)hyper)[t];
  dst[t] = h.x * m0 + h.y * m1 + h.z * m2 + h.w * m3;
}

// ------- generic projection: Y[n,j] = sum_i X[bmap(n),i] * W[j*256+i] -------
// one wave = one 16x16 output tile; 8 waves / block; ROWS x 256 output.
__global__ __launch_bounds__(256) void gemm256_kernel(
    const float* __restrict__ X, const float* __restrict__ W,
    float* __restrict__ Y, int xRowsPerB, int xRowOff) {
  const int lane = threadIdx.x & 31;
  const int wave = threadIdx.x >> 5;
  const int tile = blockIdx.x * 8 + wave;   // (ROWS/16)*16 = 25600 tiles
  const int n0 = (tile >> 4) * 16;
  const int j0 = (tile & 15) * 16;
  const int r = lane & 15, half = lane >> 4, kb = half * 2;
  const int nIn = n0 + r;
  const int bb = nIn / POMO, pp = nIn % POMO;
  const float* xrow = X + ((size_t)bb * xRowsPerB + xRowOff + pp) * 256;
  const float* wrow = W + (size_t)(j0 + r) * 256;
  v8f c = {};
#pragma unroll 8
  for (int k = 0; k < 256; k += 4) {
    v2f a = *(const v2f*)(xrow + k + kb);
    v2f b = *(const v2f*)(wrow + k + kb);
    c = wmma4(a, b, c);
  }
#pragma unroll
  for (int v = 0; v < 8; ++v)
    Y[(size_t)(n0 + v + 8 * half) * 256 + j0 + r] = c[v];
}

// ------- q projection: K = 257 (concat last col = load), W stride 257 -------
__global__ __launch_bounds__(256) void qproj_kernel(
    const float* __restrict__ eln, const float* __restrict__ loadv,
    const float* __restrict__ Wq, float* __restrict__ Q) {
  const int lane = threadIdx.x & 31;
  const int wave = threadIdx.x >> 5;
  const int tile = blockIdx.x * 8 + wave;
  const int n0 = (tile >> 4) * 16;
  const int j0 = (tile & 15) * 16;
  const int r = lane & 15, half = lane >> 4, kb = half * 2;
  const float* xrow = eln + (size_t)(n0 + r) * 256;
  const float* wrow = Wq + (size_t)(j0 + r) * 257;  // 4-byte aligned rows
  v8f c = {};
#pragma unroll 8
  for (int k = 0; k < 256; k += 4) {
    v2f a = *(const v2f*)(xrow + k + kb);
    v2f b;                       // stride-257 rows: scalar pair (avoid 8B align req)
    b.x = wrow[k + kb];
    b.y = wrow[k + kb + 1];
    c = wmma4(a, b, c);
  }
  const float wj = wrow[256];    // tail column i=256 for output col j0+r? no: per-lane col
  // NOTE: c[v] is D[row=v+8*half][col=lane&15]; tail uses W[(j0+col)*257+256]
  const float wcol = Wq[(size_t)(j0 + r) * 257 + 256];
  (void)wj;
#pragma unroll
  for (int v = 0; v < 8; ++v) {
    const int row = n0 + v + 8 * half;
    c[v] += loadv[row] * wcol;
    Q[(size_t)row * 256 + j0 + r] = c[v];
  }
}

// ------- dual-source flash attention; writes out_n + out_s -------
__global__ __launch_bounds__(128) void attn_kernel(
    const float* __restrict__ Q, const float* __restrict__ Kn,
    const float* __restrict__ Vn, const float* __restrict__ Ks,
    const float* __restrict__ Vs, const float* __restrict__ nmask,
    const float* __restrict__ smask, float* __restrict__ AO) {
  __shared__ float lds[4][16 * 208];
  const int lane = threadIdx.x & 31;
  const int wave = threadIdx.x >> 5;
  const int task = blockIdx.x * 4 + wave;       // BATCH*NHEAD*NTILE = 13312 exact
  const int nt = task % NTILE;
  const int h = (task / NTILE) % NHEAD;
  const int b = task / (NTILE * NHEAD);
  const int n0 = nt * 16;
  float* wl = lds[wave];
  const int r = lane & 15, half = lane >> 4, kb = half * 2;
  const int qr = (n0 + r < POMO) ? (n0 + r) : POMO - 1;
  const float* qrow = Q + ((size_t)(b * POMO + qr)) * 256 + h * HD;
  v8f o0 = {}, o1 = {};
  for (int pass = 0; pass < 2; ++pass) {
    const float* Kp = pass ? Ks : Kn;
    const float* Vp = pass ? Vs : Vn;
    const float* Mp = pass ? smask : nmask;
    // ---- scores (16 x 208) into LDS ----
    for (int mt = 0; mt < NTILE; ++mt) {
      const int m0 = mt * 16;
      const int krr = (m0 + r < NODE) ? (m0 + r) : NODE - 1;
      const float* krow = Kp + ((size_t)(b * NODE + krr)) * 256 + h * HD;
      v8f s = {};
#pragma unroll
      for (int d = 0; d < HD; d += 4) {
        v2f a = *(const v2f*)(qrow + d + kb);
        a *= 0.17677669529663687f;  // 1/sqrt(32)
        v2f bv = *(const v2f*)(krow + d + kb);
        s = wmma4(a, bv, s);
      }
      const int m = m0 + r;
#pragma unroll
      for (int v = 0; v < 8; ++v) {
        const int row = v + 8 * half;
        const int n = n0 + row;
        float val = -INFINITY;
        if (n < POMO && m < NODE)
          val = s[v] + Mp[((size_t)b * POMO + n) * NODE + m];
        wl[row * 208 + m0 + r] = val;
      }
    }
    asm volatile("s_wait_dscnt 0" ::: "memory");
    // ---- per-row softmax over 208 (padded -inf) ----
    if (lane < 16) {
      float* rowp = wl + lane * 208;
      float mx = -INFINITY;
      for (int m = 0; m < 208; ++m) mx = fmaxf(mx, rowp[m]);
      if (mx > -INFINITY) {
        float sum = 0.f;
        for (int m = 0; m < 208; ++m) {
          float e = __expf(rowp[m] - mx);
          rowp[m] = e;
          sum += e;
        }
        const float inv = 1.f / sum;
        for (int m = 0; m < 208; ++m) rowp[m] *= inv;
      } else {
        for (int m = 0; m < 208; ++m) rowp[m] = 0.f;
      }
    }
    asm volatile("s_wait_dscnt 0" ::: "memory");
    // ---- out += P @ V (accumulates both passes) ----
    for (int m = 0; m < 208; m += 4) {
      v2f a;
      a.x = wl[r * 208 + m + kb];
      a.y = wl[r * 208 + m + kb + 1];
      const int m0c = (m + kb < NODE) ? m + kb : NODE - 1;       // prob==0 on pad
      const int m1c = (m + kb + 1 < NODE) ? m + kb + 1 : NODE - 1;
      const float* v0 = Vp + ((size_t)(b * NODE + m0c)) * 256 + h * HD;
      const float* v1 = Vp + ((size_t)(b * NODE + m1c)) * 256 + h * HD;
      v2f b0, b1;
      b0.x = v0[r];      b0.y = v1[r];
      b1.x = v0[16 + r]; b1.y = v1[16 + r];
      o0 = wmma4(a, b0, o0);
      o1 = wmma4(a, b1, o1);
    }
  }
#pragma unroll
  for (int v = 0; v < 8; ++v) {
    const int n = n0 + v + 8 * half;
    if (n < POMO) {
      float* dst = AO + ((size_t)(b * POMO + n)) * 256 + h * HD;
      dst[r] = o0[v];
      dst[16 + r] = o1[v];
    }
  }
}

// ------- final: probs = softmax(10*tanh((mh . nodes^T)/16) + ninf_mask) -------
__global__ __launch_bounds__(128) void logits_kernel(
    const float* __restrict__ MH, const float* __restrict__ enc,
    const float* __restrict__ nmask, float* __restrict__ out) {
  __shared__ float lds[4][16 * 208];
  const int lane = threadIdx.x & 31;
  const int wave = threadIdx.x >> 5;
  const int task = blockIdx.x * 4 + wave;       // BATCH*NTILE = 1664 exact
  const int nt = task % NTILE;
  const int b = task / NTILE;
  const int n0 = nt * 16;
  float* wl = lds[wave];
  const int r = lane & 15, half = lane >> 4, kb = half * 2;
  const int qr = (n0 + r < POMO) ? (n0 + r) : POMO - 1;
  const float* mrow = MH + ((size_t)(b * POMO + qr)) * 256;
  for (int mt = 0; mt < NTILE; ++mt) {
    const int m0 = mt * 16;
    const int nr = (m0 + r < NODE) ? (m0 + r) : NODE - 1;
    const float* nrow = enc + ((size_t)b * (NODE + SOL) + nr) * 256;
    v8f s = {};
#pragma unroll 8
    for (int e = 0; e < 256; e += 4) {
      v2f a = *(const v2f*)(mrow + e + kb);
      v2f bv = *(const v2f*)(nrow + e + kb);
      s = wmma4(a, bv, s);
    }
    const int m = m0 + r;
#pragma unroll
    for (int v = 0; v < 8; ++v) {
      const int row = v + 8 * half;
      const int n = n0 + row;
      float val = -INFINITY;
      if (n < POMO && m < NODE)
        val = 10.f * tanhf(s[v] * 0.0625f) +
              nmask[((size_t)b * POMO + n) * NODE + m];
      wl[row * 208 + m0 + r] = val;
    }
  }
  asm volatile("s_wait_dscnt 0" ::: "memory");
  if (lane < 16) {
    float* rowp = wl + lane * 208;
    float mx = -INFINITY;
    for (int m = 0; m < 208; ++m) mx = fmaxf(mx, rowp[m]);
    if (mx > -INFINITY) {
      float sum = 0.f;
      for (int m = 0; m < 208; ++m) {
        float e = __expf(rowp[m] - mx);
        rowp[m] = e;
        sum += e;
      }
      const float inv = 1.f / sum;
      for (int m = 0; m < 208; ++m) rowp[m] *= inv;
    }
  }
  asm volatile("s_wait_dscnt 0" ::: "memory");
  for (int row = 0; row < 16; ++row) {
    const int n = n0 + row;
    if (n >= POMO) break;  // uniform across wave
    for (int m = lane; m < NODE; m += 32)
      out[((size_t)b * POMO + n) * NODE + m] = wl[row * 208 + m];
  }
}

extern "C" void kernel_launch(void* const* d_in, const int* in_sizes, int n_in,
                              void* d_out, int out_size, void* d_ws,
                              size_t ws_size, hipStream_t stream) {
  (void)in_sizes; (void)n_in; (void)out_size; (void)ws_size;
  const float* pref  = (const float*)d_in[0];
  const float* enc   = (const float*)d_in[1];
  const float* eln   = (const float*)d_in[2];
  const float* loadv = (const float*)d_in[3];
  const float* smask = (const float*)d_in[4];
  const float* nmask = (const float*)d_in[5];
  const float* fc1w  = (const float*)d_in[6];
  const float* fc1b  = (const float*)d_in[7];
  const float* fc2w  = (const float*)d_in[8];
  const float* fc2b  = (const float*)d_in[9];
  const float* fc3w  = (const float*)d_in[10];
  const float* fc3b  = (const float*)d_in[11];
  const float* wqh   = (const float*)d_in[12];
  const float* wkh   = (const float*)d_in[13];
  const float* wvh   = (const float*)d_in[14];
  const float* wch   = (const float*)d_in[15];
  const float* wksh  = (const float*)d_in[16];
  const float* wvsh  = (const float*)d_in[17];
  float* ws  = (float*)d_ws;
  float* out = (float*)d_out;

  float* mid = ws + OFF_MID;
  float* WQ  = ws + OFF_WQ;
  float* WK  = ws + OFF_WK;
  float* WV  = ws + OFF_WV;
  float* WC  = ws + OFF_WC;
  float* WKS = ws + OFF_WKS;
  float* WVS = ws + OFF_WVS;
  float* Qb  = ws + OFF_Q;
  float* Kb  = ws + OFF_K;
  float* Vb  = ws + OFF_V;
  float* KSb = ws + OFF_KS;
  float* VSb = ws + OFF_VS;
  float* AO  = ws + OFF_AO;
  float* MHb = ws + OFF_MH;

  mlp_kernel<<<1, 256, 0, stream>>>(pref, fc1w, fc1b, fc2w, fc2b, fc3w, fc3b, mid);

  genw_kernel<<<257, 256, 0, stream>>>(wqh,  mid, 0, WQ,  65792);
  genw_kernel<<<256, 256, 0, stream>>>(wkh,  mid, 1, WK,  65536);
  genw_kernel<<<256, 256, 0, stream>>>(wvh,  mid, 2, WV,  65536);
  genw_kernel<<<256, 256, 0, stream>>>(wch,  mid, 3, WC,  65536);
  genw_kernel<<<256, 256, 0, stream>>>(wksh, mid, 4, WKS, 65536);
  genw_kernel<<<256, 256, 0, stream>>>(wvsh, mid, 5, WVS, 65536);

  const int gemmBlocks = (ROWS / 16) * 16 / 8;  // 3200
  qproj_kernel<<<gemmBlocks, 256, 0, stream>>>(eln, loadv, WQ, Qb);
  gemm256_kernel<<<gemmBlocks, 256, 0, stream>>>(enc, WK,  Kb,  NODE + SOL, 0);
  gemm256_kernel<<<gemmBlocks, 256, 0, stream>>>(enc, WV,  Vb,  NODE + SOL, 0);
  gemm256_kernel<<<gemmBlocks, 256, 0, stream>>>(enc, WKS, KSb, NODE + SOL, NODE);
  gemm256_kernel<<<gemmBlocks, 256, 0, stream>>>(enc, WVS, VSb, NODE + SOL, NODE);

  attn_kernel<<<(BATCH * NHEAD * NTILE) / 4, 128, 0, stream>>>(
      Qb, Kb, Vb, KSb, VSb, nmask, smask, AO);

  gemm256_kernel<<<gemmBlocks, 256, 0, stream>>>(AO, WC, MHb, POMO, 0);

  logits_kernel<<<(BATCH * NTILE) / 4, 128, 0, stream>>>(MHb, enc, nmask, out);
}